// Encoder_44985487458945
// MI455X (gfx1250) — compile-verified
//
#include <hip/hip_runtime.h>

#define L      4096
#define CH     512          // positions per workgroup
#define CHUNKS (L / CH)     // 8
#define BATCH  128
#define DIM    64
#define EPS    1e-3f

typedef float v2f __attribute__((ext_vector_type(2)));
typedef float v8f __attribute__((ext_vector_type(8)));

struct EncParams {
    const float* x;            // [B, L, 1]
    const int*   perm;         // [B, L]
    const float* cw[3];        // [5,1,64]
    const float* cb[3];        // [64]
    const float* cg[3];
    const float* cbe[3];
    const float* cm[3];
    const float* cv[3];
    const float* fw[3];        // [64,1]
    const float* fb[3];        // [1]
    const float* mpw;          // [1,64]
    const float* mpb; const float* mpg; const float* mpbe; const float* mpm; const float* mpv;
    const float* mdw;          // [64,2]
    const float* mdb; const float* mdg; const float* mdbe; const float* mdm; const float* mdv;
    float*       out;          // [B, 3L, 2]
};

__device__ __forceinline__ float elu1(float z) {
    return z > 0.0f ? z : (__expf(z) - 1.0f);
}

__global__ __launch_bounds__(256) void encoder_fused(EncParams P) {
    __shared__ float  row_s[L];          // 16 KB: one input row, +/-1 mapped
    __shared__ float  xI_s[CH + 8];      // gathered (interleaved) window incl. halo
    __shared__ float  e_s[3 * CH];       // per-branch encoder scalars
    __shared__ float  wS_s[3 * 5 * 64];  // BN-folded conv weights
    __shared__ float  beta_s[3 * 64];    // BN-folded conv bias
    __shared__ float  fwv_s[3 * 64];     // projection weights
    __shared__ float  fb_s[4];           // projection biases
    __shared__ float4 mc_s[64];          // modulation {A, C, MD0, MD1}
    __shared__ float  kk_s[2];           // modulation output offsets

    const int tid  = threadIdx.x;
    const int b    = blockIdx.y;
    const int l0   = blockIdx.x * CH;    // chunk start

    // ---- fold constants into LDS (runs in parallel with row load) ----
    if (tid < 192) {
        const int p = tid >> 6, d = tid & 63;
        const float s = P.cg[p][d] * rsqrtf(P.cv[p][d] + EPS);
        beta_s[p * 64 + d] = s * (P.cb[p][d] - P.cm[p][d]) + P.cbe[p][d];
#pragma unroll
        for (int k = 0; k < 5; ++k)
            wS_s[p * 320 + k * 64 + d] = s * P.cw[p][k * 64 + d];
        fwv_s[p * 64 + d] = P.fw[p][d];
        if (d == 0) fb_s[p] = P.fb[p][0];
    } else {
        const int d = tid - 192;
        const float s2  = P.mpg[d] * rsqrtf(P.mpv[d] + EPS);
        const float s30 = P.mdg[0] * rsqrtf(P.mdv[0] + EPS);
        const float s31 = P.mdg[1] * rsqrtf(P.mdv[1] + EPS);
        mc_s[d] = make_float4(s2 * P.mpw[d],
                              s2 * (P.mpb[d] - P.mpm[d]) + P.mpbe[d],
                              s30 * P.mdw[d * 2 + 0],
                              s31 * P.mdw[d * 2 + 1]);
        if (d == 0) {
            kk_s[0] = s30 * (P.mdb[0] - P.mdm[0]) + P.mdbe[0];
            kk_s[1] = s31 * (P.mdb[1] - P.mdm[1]) + P.mdbe[1];
        }
    }

    // ---- load full input row into LDS (bits -> +/-1) ----
    const float* xrow = P.x + (size_t)b * L;
    for (int i = tid; i < L; i += 256)
        row_s[i] = 2.0f * xrow[i] - 1.0f;
    __syncthreads();

    // ---- gather interleaved window: xI_s[i] = x[b, perm[b, l0-2+i]] ----
    const int* prow = P.perm + (size_t)b * L;
    for (int i = tid; i < CH + 4; i += 256) {
        const int gl = (l0 - 2 + i + L) & (L - 1);
        xI_s[i] = row_s[prow[gl]];
    }
    __syncthreads();

    // ================= Phase 1: conv branches via V_WMMA_F32_16X16X4_F32 ====
    const int wave = tid >> 5;
    const int lane = tid & 31;
    const int half = lane >> 4;    // which 16-lane half
    const int mn   = lane & 15;    // M (for A) / N (for B,C,D)
    const int kb   = half * 2;     // K base held by this half (ISA A/B layout)

    // preload B matrices: (K=kb,kb+1 rows) x (N=mn) for 3 branches x 4 ch-tiles
    float bx[3][4], by[3][4];
#pragma unroll
    for (int p = 0; p < 3; ++p)
#pragma unroll
        for (int t = 0; t < 4; ++t) {
            bx[p][t] = wS_s[p * 320 + kb * 64 + t * 16 + mn];
            by[p][t] = wS_s[p * 320 + (kb + 1) * 64 + t * 16 + mn];
        }

    for (int T = wave * 4; T < wave * 4 + 4; ++T) {   // 16-position tiles
        const int base = l0 + T * 16;
        // A matrices (16x4): element (M=mn, K=kb..kb+1) = x[base + M + K - 2]
        v2f aS, aI;
        aS.x = row_s[(base + mn + kb - 2 + L) & (L - 1)];
        aS.y = row_s[(base + mn + kb - 1 + L) & (L - 1)];
        aI.x = xI_s[T * 16 + mn + kb];
        aI.y = xI_s[T * 16 + mn + kb + 1];

#pragma unroll
        for (int p = 0; p < 3; ++p) {
            const v2f a = (p == 2) ? aI : aS;
            float acc[8];
#pragma unroll
            for (int r = 0; r < 8; ++r) acc[r] = 0.0f;

#pragma unroll
            for (int t = 0; t < 4; ++t) {
                v2f bm; bm.x = bx[p][t]; bm.y = by[p][t];
                v8f c = {};
                v8f dm = __builtin_amdgcn_wmma_f32_16x16x4_f32(
                    false, a, false, bm, (short)0, c, false, false);
                const int d0 = t * 16;
#pragma unroll
                for (int r = 0; r < 8; ++r) {
                    const int rowi = r + half * 8;     // C/D row for this lane
                    const float x5 = (p == 2)
                        ? xI_s[T * 16 + rowi + 4]
                        : row_s[(base + rowi + 2) & (L - 1)];
                    float z = dm[r] + wS_s[p * 320 + 256 + d0 + mn] * x5
                                    + beta_s[p * 64 + d0 + mn];
                    acc[r] += fwv_s[p * 64 + d0 + mn] * elu1(z);
                }
            }
            // reduce over 16 columns (lanes within each half), store e
#pragma unroll
            for (int r = 0; r < 8; ++r) {
                float s = acc[r];
                s += __shfl_xor(s, 1, 16);
                s += __shfl_xor(s, 2, 16);
                s += __shfl_xor(s, 4, 16);
                s += __shfl_xor(s, 8, 16);
                if (mn == 0)
                    e_s[p * CH + T * 16 + r + half * 8] = s + fb_s[p];
            }
        }
    }
    __syncthreads();

    // ================= Phase 2: pointwise modulation, coalesced f2 stores ===
    const float K0 = kk_s[0], K1 = kk_s[1];
    float2* out2 = (float2*)P.out + (size_t)b * (3 * L) + 3 * l0;
#pragma unroll
    for (int k = 0; k < 6; ++k) {
        const int tl = tid + k * 256;          // 0 .. 3*CH-1, t = 3q + p
        const float e = e_s[(tl % 3) * CH + (tl / 3)];
        float a0 = K0, a1 = K1;
#pragma unroll
        for (int d = 0; d < 64; ++d) {
            const float4 m = mc_s[d];
            const float mo = elu1(m.x * e + m.y);
            a0 += m.z * mo;
            a1 += m.w * mo;
        }
        out2[tl] = make_float2(a0, a1);
    }
}

extern "C" void kernel_launch(void* const* d_in, const int* in_sizes, int n_in,
                              void* d_out, int out_size, void* d_ws, size_t ws_size,
                              hipStream_t stream) {
    (void)in_sizes; (void)n_in; (void)out_size; (void)d_ws; (void)ws_size;
    EncParams P;
    P.x    = (const float*)d_in[0];
    P.perm = (const int*)  d_in[1];
    for (int p = 0; p < 3; ++p) {
        const int base = 2 + 6 * p;            // c1 / c2 / cI blocks
        P.cw[p]  = (const float*)d_in[base + 0];
        P.cb[p]  = (const float*)d_in[base + 1];
        P.cg[p]  = (const float*)d_in[base + 2];
        P.cbe[p] = (const float*)d_in[base + 3];
        P.cm[p]  = (const float*)d_in[base + 4];
        P.cv[p]  = (const float*)d_in[base + 5];
    }
    P.fw[0] = (const float*)d_in[20]; P.fb[0] = (const float*)d_in[21];
    P.fw[1] = (const float*)d_in[22]; P.fb[1] = (const float*)d_in[23];
    P.fw[2] = (const float*)d_in[24]; P.fb[2] = (const float*)d_in[25];
    P.mpw  = (const float*)d_in[26]; P.mpb  = (const float*)d_in[27];
    P.mpg  = (const float*)d_in[28]; P.mpbe = (const float*)d_in[29];
    P.mpm  = (const float*)d_in[30]; P.mpv  = (const float*)d_in[31];
    P.mdw  = (const float*)d_in[32]; P.mdb  = (const float*)d_in[33];
    P.mdg  = (const float*)d_in[34]; P.mdbe = (const float*)d_in[35];
    P.mdm  = (const float*)d_in[36]; P.mdv  = (const float*)d_in[37];
    P.out  = (float*)d_out;

    dim3 grid(CHUNKS, BATCH);   // 8 x 128 workgroups
    encoder_fused<<<grid, 256, 0, stream>>>(P);
}